// SemanticGridPoolingLayer_20083267076195
// MI455X (gfx1250) — compile-verified
//
#include <hip/hip_runtime.h>
#include <hip/hip_bf16.h>
#include <math.h>

typedef __attribute__((ext_vector_type(2))) float v2f;
typedef __attribute__((ext_vector_type(8))) float v8f;

#define IN_C  96
#define OUT_C 192
#define NB    4
#define EPSV  1e-5f

// ---------------- zero workspace ----------------
__global__ void k_zero(float* ws, long long nwords) {
    long long t = (long long)blockIdx.x * blockDim.x + threadIdx.x;
    if (t < nwords) ws[t] = 0.0f;
}

// ---------------- segment-sum scatter (one block of 96 threads per point) ----------------
__global__ void k_scatter(const float* __restrict__ feat,
                          const float* __restrict__ scores,
                          const int*   __restrict__ cluster,
                          float* __restrict__ pooled,
                          float* __restrict__ cnt,
                          float* __restrict__ ssum,
                          int N) {
    int i = blockIdx.x;
    if (i >= N) return;
    int g = cluster[i];
    int c = threadIdx.x;
    unsafeAtomicAdd(&pooled[(long long)g * IN_C + c], feat[(long long)i * IN_C + c]);
    if (c == 0) {
        unsafeAtomicAdd(&cnt[g], 1.0f);
        unsafeAtomicAdd(&ssum[g], scores[i]);
    }
}

// ---------------- divide sums by counts; emit pooled means + new_scores ----------------
__global__ void k_finalize(float* __restrict__ pooled,
                           const float* __restrict__ cnt,
                           const float* __restrict__ ssum,
                           float* __restrict__ out_scores,
                           int G) {
    long long t = (long long)blockIdx.x * blockDim.x + threadIdx.x;
    if (t >= (long long)G * IN_C) return;
    int g = (int)(t / IN_C);
    int c = (int)(t % IN_C);
    float inv = 1.0f / fmaxf(cnt[g], 1.0f);
    pooled[t] *= inv;
    if (c == 0) out_scores[g] = ssum[g] * inv;
}

// ---------------- fused fp32 WMMA GEMM + LayerNorm + exact GELU ----------------
// one wave per 16-row block; 12 accumulator tiles cover OUT_C=192.
// Per k-step: one 8B A load (branchless boundary zeroing via select) + twelve 8B
// B loads into distinct registers, then 12 WMMAs -> spill-free clause pipelining.
__global__ void k_gemm_ln_gelu(const float* __restrict__ pooled,
                               const float* __restrict__ Wm,
                               const float* __restrict__ gamma,
                               const float* __restrict__ beta,
                               float* __restrict__ out_feat,
                               int G) {
    int wave = threadIdx.x >> 5;
    int lane = threadIdx.x & 31;
    int rb   = blockIdx.x * (blockDim.x >> 5) + wave;
    int m0   = rb * 16;
    if (m0 >= G) return;                 // wave-uniform exit

    int half = lane >> 4;                // 0 or 1
    int lm   = lane & 15;

    // A lane l, vgpr v holds A[M = l&15][K = 2*(l>>4) + v]
    int  arow   = m0 + lm;
    bool avalid = arow < G;
    int  safer  = avalid ? arow : (G - 1);          // clamp: always a legal address
    const float* ap = pooled + (long long)safer * IN_C + 2 * half;
    float amask = avalid ? 1.0f : 0.0f;             // branchless zeroing

    // B lane l, vgpr v holds B[K = 2*(l>>4) + v][N = l&15] = W[n][k]
    const float* wbase = Wm + (long long)lm * IN_C + 2 * half;

    v8f acc[12] = {};

    for (int ks = 0; ks < IN_C / 4; ++ks) {
        v2f a = *(const v2f*)(ap + 4 * ks);
        a.x *= amask;
        a.y *= amask;
        v2f b[12];
#pragma unroll
        for (int t = 0; t < 12; ++t) {
            b[t] = *(const v2f*)(wbase + (long long)t * 16 * IN_C + 4 * ks);
        }
#pragma unroll
        for (int t = 0; t < 12; ++t) {
            acc[t] = __builtin_amdgcn_wmma_f32_16x16x4_f32(
                false, a, false, b[t], (short)0, acc[t], false, false);
        }
    }

    // per-row LayerNorm stats: row m = v + 8*half lives in one 16-lane half
    float mean_a[8], rsig_a[8];
#pragma unroll
    for (int v = 0; v < 8; ++v) {
        float s = 0.0f, s2 = 0.0f;
#pragma unroll
        for (int t = 0; t < 12; ++t) { float x = acc[t][v]; s += x; s2 += x * x; }
#pragma unroll
        for (int off = 1; off < 16; off <<= 1) {
            s  += __shfl_xor(s,  off, 32);
            s2 += __shfl_xor(s2, off, 32);
        }
        float mean = s * (1.0f / OUT_C);
        float var  = s2 * (1.0f / OUT_C) - mean * mean;
        mean_a[v] = mean;
        rsig_a[v] = rsqrtf(var + EPSV);
    }

#pragma unroll
    for (int t = 0; t < 12; ++t) {
        int n = t * 16 + lm;
        float ga = gamma[n], be = beta[n];
#pragma unroll
        for (int v = 0; v < 8; ++v) {
            int row = m0 + v + 8 * half;
            if (row < G) {
                float x = (acc[t][v] - mean_a[v]) * rsig_a[v] * ga + be;
                float y = 0.5f * x * (1.0f + erff(x * 0.70710678118f));
                out_feat[(long long)row * OUT_C + n] = y;
            }
        }
    }
}

// ---------------- gather side outputs + per-batch histogram ----------------
__global__ void k_gather(const int*   __restrict__ grid_coord,
                         const float* __restrict__ coord,
                         const int*   __restrict__ batch,
                         const int*   __restrict__ unique_idx,
                         int* __restrict__ out_grid,
                         float* __restrict__ out_coord,
                         int* __restrict__ out_batch,
                         long long* __restrict__ out_code,
                         long long* __restrict__ out_ord1,
                         long long* __restrict__ out_ord2,
                         int* __restrict__ per_batch,
                         int G) {
    int g = blockIdx.x * blockDim.x + threadIdx.x;
    if (g >= G) return;
    int gi = unique_idx[g];
#pragma unroll
    for (int j = 0; j < 3; ++j) {
        out_grid[g * 3 + j]  = grid_coord[gi * 3 + j] >> 1;   // // STRIDE(2), coords >= 0
        out_coord[g * 3 + j] = coord[gi * 3 + j];
    }
    int b = batch[gi];
    out_batch[g] = b;
    atomicAdd(&per_batch[b], 1);
    out_code[g] = ((long long)b << 30) + (long long)g;        // CODE_DEPTH*3 = 30
    out_ord1[g] = (long long)g;
    out_ord2[g] = (long long)g;
}

// ---------------- 4-element cumsum for offsets ----------------
__global__ void k_offsets(const int* __restrict__ per_batch, int* __restrict__ out_offset) {
    if (blockIdx.x == 0 && threadIdx.x == 0) {
        int s = 0;
#pragma unroll
        for (int b = 0; b < NB; ++b) { s += per_batch[b]; out_offset[b] = s; }
    }
}

extern "C" void kernel_launch(void* const* d_in, const int* in_sizes, int n_in,
                              void* d_out, int out_size, void* d_ws, size_t ws_size,
                              hipStream_t stream) {
    const float* feat       = (const float*)d_in[0];
    const float* scores     = (const float*)d_in[1];
    const float* coord      = (const float*)d_in[2];
    const int*   grid_coord = (const int*)  d_in[3];
    const int*   batch      = (const int*)  d_in[4];
    const int*   cluster    = (const int*)  d_in[5];
    const int*   unique_idx = (const int*)  d_in[6];
    const float* Wm         = (const float*)d_in[7];
    const float* gamma      = (const float*)d_in[8];
    const float* beta       = (const float*)d_in[9];

    int N = in_sizes[5];          // cluster_idx has N elements
    int G = in_sizes[6];          // unique_idx has G elements

    // output tuple flattened in return order
    float*     out_feat   = (float*)d_out;                       // G*192 f32
    float*     out_scores = out_feat + (size_t)G * OUT_C;        // G f32
    int*       out_grid   = (int*)(out_scores + G);              // 3G i32
    float*     out_coord  = (float*)(out_grid + (size_t)3 * G);  // 3G f32
    int*       out_batch  = (int*)(out_coord + (size_t)3 * G);   // G i32
    int*       out_offset = out_batch + G;                       // 4 i32
    long long* out_code   = (long long*)(out_offset + NB);       // G i64 (8B-aligned)
    long long* out_ord1   = out_code + G;                        // G i64
    long long* out_ord2   = out_ord1 + G;                        // G i64

    // workspace
    float* pooled    = (float*)d_ws;                             // G*96
    float* cnt       = pooled + (size_t)G * IN_C;                // G
    float* ssum      = cnt + G;                                  // G
    int*   per_batch = (int*)(ssum + G);                         // 4

    long long nwords = (long long)G * IN_C + 2LL * G + NB;
    k_zero<<<(int)((nwords + 255) / 256), 256, 0, stream>>>((float*)d_ws, nwords);

    k_scatter<<<N, IN_C, 0, stream>>>(feat, scores, cluster, pooled, cnt, ssum, N);

    long long nf = (long long)G * IN_C;
    k_finalize<<<(int)((nf + 255) / 256), 256, 0, stream>>>(pooled, cnt, ssum, out_scores, G);

    int row_blocks = (G + 15) / 16;
    k_gemm_ln_gelu<<<(row_blocks + 7) / 8, 256, 0, stream>>>(pooled, Wm, gamma, beta, out_feat, G);

    k_gather<<<(G + 255) / 256, 256, 0, stream>>>(grid_coord, coord, batch, unique_idx,
                                                  out_grid, out_coord, out_batch,
                                                  out_code, out_ord1, out_ord2, per_batch, G);

    k_offsets<<<1, 64, 0, stream>>>(per_batch, out_offset);
}